// ColorEntropyLoss_90915867722048
// MI455X (gfx1250) — compile-verified
//
#include <hip/hip_runtime.h>
#include <hip/hip_bf16.h>
#include <math.h>

// ColorEntropyLoss on MI455X (gfx1250, wave32).
// Batched GEMM formulation: color_dist[b] = attn[b] (16x4096) x onehot(grids[b]) (4096x16pad)
// using V_WMMA_F32_16X16X4_F32 with full f32 accumulation (exact vs reference math).

typedef __attribute__((ext_vector_type(2))) float v2f;
typedef __attribute__((ext_vector_type(8))) float v8f;

#define BATCH      512
#define SROWS      16
#define HW         4096
#define NUM_COLORS 10
#define EPS        1e-8f

__global__ __launch_bounds__(256)
void color_entropy_partial(const float* __restrict__ attn,
                           const int*   __restrict__ grids,
                           float*       __restrict__ partial) {
    const int b    = blockIdx.x;
    const int tid  = threadIdx.x;
    const int wave = tid >> 5;
    const int lane = tid & 31;

    const float* attnb = attn  + (size_t)b * SROWS * HW;
    const int*   gridb = grids + (size_t)b * HW;

    // WMMA f32 16x16x4 fragment geometry (wave32):
    //   A (16x4):  lanes 0-15 -> M=lane,  VGPR0/1 = K0/K1; lanes 16-31 -> M=lane-16, K2/K3
    //   B (4x16):  VGPR v, lane group g -> row K = v + 2*g, col N = lane&15
    const int row  = lane & 15;          // M (s-row) for A, N (color col) for B
    const int koff = (lane >> 4) << 1;   // 0 or 2: this lane's K sub-offset
    const int col  = lane & 15;

    v8f acc = (v8f){0.f, 0.f, 0.f, 0.f, 0.f, 0.f, 0.f, 0.f};

    const int kchunk = HW / 8;           // 512 K-values per wave
    const int ksteps = kchunk / 4;       // 128 WMMA steps per wave
    const int kbase  = wave * kchunk;

    const float* aptr = attnb + (size_t)row * HW + kbase + koff;
    const int*   gptr = gridb + kbase + koff;

    for (int i = 0; i < ksteps; ++i) {
        v2f a;
        a.x = aptr[0];
        a.y = aptr[1];
        const int g0 = gptr[0];
        const int g1 = gptr[1];
        v2f bb;
        bb.x = (g0 == col) ? 1.0f : 0.0f;
        bb.y = (g1 == col) ? 1.0f : 0.0f;
        // D = A x B + C  (f32, 16x16x4) -> v_wmma_f32_16x16x4_f32
        acc = __builtin_amdgcn_wmma_f32_16x16x4_f32(
                  /*neg_a=*/false, a, /*neg_b=*/false, bb,
                  /*c_mod=*/(short)0, acc, /*reuse_a=*/false, /*reuse_b=*/false);
        aptr += 4;
        gptr += 4;
    }

    // ---- reduce the 8 per-wave accumulators via LDS ----
    __shared__ float red[8 * 256];
    __shared__ float mat[16 * 16];
    __shared__ float ent[16];

    #pragma unroll
    for (int v = 0; v < 8; ++v)
        red[wave * 256 + v * 32 + lane] = acc[v];
    __syncthreads();

    {
        float s = 0.f;
        #pragma unroll
        for (int w = 0; w < 8; ++w) s += red[w * 256 + tid];
        // C/D layout: VGPR v, lane l -> M = v + (l>=16 ? 8 : 0), N = l & 15
        const int v = tid >> 5;
        const int l = tid & 31;
        const int M = v + ((l >= 16) ? 8 : 0);
        const int N = l & 15;
        mat[M * 16 + N] = s;
    }
    __syncthreads();

    // ---- per-s normalization + entropy (rows of the 16x16 result) ----
    if (tid < SROWS) {
        float rowsum = 0.f;
        #pragma unroll
        for (int c = 0; c < NUM_COLORS; ++c) rowsum += mat[tid * 16 + c];
        const float inv = 1.0f / (rowsum + EPS);
        float e = 0.f;
        #pragma unroll
        for (int c = 0; c < NUM_COLORS; ++c) {
            const float p = mat[tid * 16 + c] * inv;
            e -= p * logf(p + EPS);
        }
        ent[tid] = e;
    }
    __syncthreads();

    if (tid == 0) {
        float s = 0.f;
        #pragma unroll
        for (int i = 0; i < SROWS; ++i) s += ent[i];
        partial[b] = s;   // sum of 16 entropies for this batch
    }
}

// Deterministic final reduction: 512 partials -> mean over B*S.
__global__ __launch_bounds__(512)
void color_entropy_finalize(const float* __restrict__ partial,
                            float*       __restrict__ out) {
    __shared__ float buf[512];
    const int tid = threadIdx.x;
    buf[tid] = partial[tid];
    __syncthreads();
    for (int s = 256; s > 0; s >>= 1) {
        if (tid < s) buf[tid] += buf[tid + s];
        __syncthreads();
    }
    if (tid == 0) out[0] = buf[0] / (float)(BATCH * SROWS);
}

extern "C" void kernel_launch(void* const* d_in, const int* in_sizes, int n_in,
                              void* d_out, int out_size, void* d_ws, size_t ws_size,
                              hipStream_t stream) {
    const float* attn  = (const float*)d_in[0];   // [512,16,64,64] f32
    const int*   grids = (const int*)d_in[1];     // [512,64,64] int
    float* out     = (float*)d_out;               // scalar f32
    float* partial = (float*)d_ws;                // 512 floats of scratch

    (void)in_sizes; (void)n_in; (void)out_size; (void)ws_size;

    color_entropy_partial<<<BATCH, 256, 0, stream>>>(attn, grids, partial);
    color_entropy_finalize<<<1, 512, 0, stream>>>(partial, out);
}